// DynamicRouting_704374636683
// MI455X (gfx1250) — compile-verified
//
#include <hip/hip_runtime.h>
#include <math.h>

typedef __attribute__((ext_vector_type(2))) float        v2f;
typedef __attribute__((ext_vector_type(8))) float        v8f;
typedef __attribute__((ext_vector_type(4))) unsigned int v4u;
typedef __attribute__((ext_vector_type(8))) int          v8i;
typedef __attribute__((ext_vector_type(4))) int          v4i;

#define CAPS_EPS 1e-8f
#define NJ 10
#define NI 1152
#define NN 16
#define NTHREADS 320   // 10 wave32s: one wave per parent capsule j
#define CH 48          // i-rows per TDM chunk (3 KB)
#define NCH (NI / CH)  // 24 chunks

// Issue one Tensor-Data-Mover DMA: nelem contiguous f32 from global -> LDS.
// D# per CDNA5 ISA ch.8: group0 = {count, lds_addr, global_addr, type=2},
// group1 = {data_size=4B, tensor_dim0, tensor_dim1=1, tile_dim0=nelem,
//           tile_dim1=1, tensor_dim0_stride}, groups 2/3 zero (<=2D tensor).
__device__ __forceinline__ void tdm_load_f32(const float* gsrc, float* ldst,
                                             unsigned nelem) {
    const unsigned lds = (unsigned)(unsigned long long)(const void*)ldst;
    const unsigned long long ga = (unsigned long long)(const void*)gsrc;
    const unsigned td0 = (unsigned)(NI * NN);   // tensor dim0 (elements)

    v4u g0;
    g0[0] = 1u;                                   // count=1, user descriptor
    g0[1] = lds;                                  // lds_addr [63:32]
    g0[2] = (unsigned)ga;                         // global_addr lo
    g0[3] = ((unsigned)(ga >> 32) & 0x01FFFFFFu)  // global_addr [56:32]
            | 0x80000000u;                        // type=2 ("image") [127:126]

    v8i g1;
    g1[0] = (int)(2u << 16);                      // data_size = 4 bytes
    g1[1] = (int)((td0 & 0xFFFFu) << 16);         // tensor_dim0[15:0] -> [63:48]
    g1[2] = (int)((td0 >> 16) | (1u << 16));      // tensor_dim0[31:16], tensor_dim1=1
    g1[3] = (int)(nelem << 16);                   // tensor_dim1 hi=0, tile_dim0=nelem
    g1[4] = 1;                                    // tile_dim1=1, tile_dim2=0
    g1[5] = (int)td0;                             // tensor_dim0_stride lo32
    g1[6] = 0;                                    // stride0 hi16, stride1 lo16
    g1[7] = 0;                                    // stride1 hi32
    const v4i z4 = {0, 0, 0, 0};
    const v8i z8 = {0, 0, 0, 0, 0, 0, 0, 0};
    __builtin_amdgcn_tensor_load_to_lds(g0, g1, z4, z4, z8, 0);
}

__global__ __launch_bounds__(NTHREADS)
void DynamicRouting_gfx1250_kernel(const float* __restrict__ u_hat,
                                   const float* __restrict__ bias,
                                   const int* __restrict__ iters_p,
                                   float* __restrict__ out)
{
    __shared__ float b_lds[NJ][NI];             // routing logits   (45 KB)
    __shared__ float c_lds[NJ][NI];             // coupling coeffs  (45 KB)
    __shared__ float stage[NJ][2][CH * NN];     // TDM double-buffer (60 KB)
    __shared__ float s_lds[NJ][NN];
    __shared__ float v_lds[NJ][NN];

    const int tid  = threadIdx.x;
    const int bi   = blockIdx.x;
    const int lane = tid & 31;
    const int wj   = tid >> 5;        // wave id == parent capsule j (0..9)
    const int hi   = (lane >> 4) & 1; // half-wave: K slots (2,3) vs (0,1)
    const int col  = lane & 15;

    const int iters = iters_p[0];
    const float* up = u_hat + ((size_t)bi * NJ + wj) * (size_t)(NI * NN);

    for (int i = tid; i < NJ * NI; i += NTHREADS) ((float*)b_lds)[i] = 0.0f;
    __syncthreads();

    const int k0 = hi ? 2 : 0;   // K slots owned by this half-wave (adjacent pair)
    const int k1 = k0 + 1;

    for (int it = 0; it < iters; ++it) {
        // ---- softmax over j: materialize c[j][i] in LDS ----
        for (int i = tid; i < NI; i += NTHREADS) {
            float mx = b_lds[0][i];
            #pragma unroll
            for (int j = 1; j < NJ; ++j) mx = fmaxf(mx, b_lds[j][i]);
            float e[NJ], sum = 0.0f;
            #pragma unroll
            for (int j = 0; j < NJ; ++j) { e[j] = __expf(b_lds[j][i] - mx); sum += e[j]; }
            const float r = 1.0f / sum;
            #pragma unroll
            for (int j = 0; j < NJ; ++j) c_lds[j][i] = e[j] * r;
        }
        __syncthreads();

        // ---- s[j][n] = sum_i c[j][i]*u[j][i][n] : TDM-staged WMMA sweep ----
        {
            v8f acc0 = 0.f, acc1 = 0.f, acc2 = 0.f, acc3 = 0.f;
            tdm_load_f32(up, stage[wj][0], CH * NN);
            for (int ch = 0; ch < NCH; ++ch) {
                if (ch + 1 < NCH) {
                    tdm_load_f32(up + (size_t)(ch + 1) * CH * NN,
                                 stage[wj][(ch + 1) & 1], CH * NN);
                    __builtin_amdgcn_s_wait_tensorcnt(1);
                } else {
                    __builtin_amdgcn_s_wait_tensorcnt(0);
                }
                const float* sb = stage[wj][ch & 1];
                const int gbase = ch * CH;
                #pragma unroll
                for (int i0 = 0; i0 < CH; i0 += 4) {
                    // A: coupling coeffs, adjacent pair -> one ds_load_b64
                    const float2 ca = *(const float2*)&c_lds[wj][gbase + i0 + k0];
                    v2f A; A[0] = ca.x; A[1] = ca.y;
                    // B[k][n] = u[i0+k][n]: two staged rows 16 floats apart
                    v2f B;
                    B[0] = sb[(i0 + k0) * NN + col];
                    B[1] = sb[(i0 + k1) * NN + col];
                    v8f& acc = ((i0 >> 2) & 3) == 0 ? acc0
                             : ((i0 >> 2) & 3) == 1 ? acc1
                             : ((i0 >> 2) & 3) == 2 ? acc2 : acc3;
                    acc = __builtin_amdgcn_wmma_f32_16x16x4_f32(
                            false, A, false, B, (short)0, acc, false, false);
                }
            }
            v8f acc = acc0 + acc1 + acc2 + acc3;  // rows replicated -> s[n]
            if (lane < 16) s_lds[wj][lane] = acc[0];
        }
        __syncthreads();

        // ---- squash ----
        if (tid < NJ * NN) {
            const int j = tid >> 4, n = tid & 15;
            float rowsum = 0.0f;
            #pragma unroll
            for (int k = 0; k < NN; ++k) rowsum += s_lds[j][k];
            float vv = 0.0f;
            if (rowsum != 0.0f) {
                float nrm = 0.0f;
                #pragma unroll
                for (int k = 0; k < NN; ++k) {
                    const float sbv = s_lds[j][k] + bias[j * NN + k];
                    nrm += sbv * sbv;
                }
                const float sbn = s_lds[j][n] + bias[j * NN + n];
                vv = (nrm / (1.0f + nrm)) * sbn * rsqrtf(nrm + CAPS_EPS);
            }
            v_lds[j][n] = vv;
        }
        __syncthreads();

        // ---- b[j][i] += sum_n u[j][i][n]*v[j][n] : TDM-staged WMMA sweep ----
        if (it < iters - 1) {
            const int m = col;  // A row index
            tdm_load_f32(up, stage[wj][0], CH * NN);
            for (int ch = 0; ch < NCH; ++ch) {
                if (ch + 1 < NCH) {
                    tdm_load_f32(up + (size_t)(ch + 1) * CH * NN,
                                 stage[wj][(ch + 1) & 1], CH * NN);
                    __builtin_amdgcn_s_wait_tensorcnt(1);
                } else {
                    __builtin_amdgcn_s_wait_tensorcnt(0);
                }
                const float* sb = stage[wj][ch & 1];
                for (int t = 0; t < CH; t += 16) {
                    v8f acc = 0.f;
                    const float* row = sb + (size_t)(t + m) * NN;
                    #pragma unroll
                    for (int n0 = 0; n0 < NN; n0 += 4) {
                        const int na = n0 + k0;
                        // A[m][k] = u[i0+m][n0+k]: adjacent pair -> ds_load_b64
                        const float2 ld = *(const float2*)&row[na];
                        v2f A; A[0] = ld.x; A[1] = ld.y;
                        // B[k][*] = v[n0+k]: adjacent pair -> ds_load_b64
                        const float2 vb = *(const float2*)&v_lds[wj][na];
                        v2f B; B[0] = vb.x; B[1] = vb.y;
                        acc = __builtin_amdgcn_wmma_f32_16x16x4_f32(
                                false, A, false, B, (short)0, acc, false, false);
                    }
                    if (col == 0) {  // lane0: rows 0..7, lane16: rows 8..15
                        const int ibase = ch * CH + t + (hi ? 8 : 0);
                        #pragma unroll
                        for (int r = 0; r < 8; ++r) b_lds[wj][ibase + r] += acc[r];
                    }
                }
            }
            __syncthreads();
        }
    }

    if (tid < NJ * NN) {
        out[(size_t)bi * (NJ * NN) + tid] = v_lds[tid >> 4][tid & 15];
    }
}

extern "C" void kernel_launch(void* const* d_in, const int* in_sizes, int n_in,
                              void* d_out, int out_size, void* d_ws, size_t ws_size,
                              hipStream_t stream) {
    (void)n_in; (void)out_size; (void)d_ws; (void)ws_size;
    const float* u_hat = (const float*)d_in[0];
    const float* bias  = (const float*)d_in[1];
    const int*   iters = (const int*)d_in[2];
    float* out = (float*)d_out;

    const int b = in_sizes[0] / (NJ * NI * NN);   // 128 for the reference shapes
    DynamicRouting_gfx1250_kernel<<<dim3(b), dim3(NTHREADS), 0, stream>>>(
        u_hat, bias, iters, out);
}